// Mix_9354438770917
// MI455X (gfx1250) — compile-verified
//
#include <hip/hip_runtime.h>

typedef float v2f __attribute__((ext_vector_type(2)));
typedef float v8f __attribute__((ext_vector_type(8)));

constexpr int BN = 2;      // batches
constexpr int CHN = 128;   // feature channels
constexpr int NP = 8192;   // points per batch
constexpr int KS = 16;     // nsample (top-k)
constexpr int WAVES = 4;   // waves per block

// One wave32 handles one tile of 16 query points: two KNN passes
// (refs = pos2 "intra", refs = pos1 "self") via V_WMMA_F32_16X16X4_F32
// Gram tiles, register-resident sorted top-16 lists, then the fused
// gather/concat of new_pos1 and new_feats1.
__global__ void __launch_bounds__(WAVES * 32)
knn_mix_kernel(const float* __restrict__ pos1, const float* __restrict__ pos2,
               const float* __restrict__ feats1, const float* __restrict__ feats2,
               const int* __restrict__ factor_p,
               float* __restrict__ out_pos, float* __restrict__ out_feats)
{
    __shared__ float s_tile[WAVES][16][32];   // dot-product tile (transposed via LDS)
    __shared__ float s_rn[WAVES][32];         // ref squared norms
    __shared__ float s_md[WAVES][32][16];     // per-lane partial top-16 dists
    __shared__ int   s_mi[WAVES][32][16];     // per-lane partial top-16 indices
    __shared__ int   s_idxS[WAVES][16][16];   // merged self-KNN indices
    __shared__ int   s_idxI[WAVES][16][16];   // merged intra-KNN indices

    const int tid  = threadIdx.x;
    const int wv   = tid >> 5;
    const int t    = tid & 31;
    const int l    = t & 15;
    const int half = t >> 4;

    const int wglobal = blockIdx.x * WAVES + wv;
    const int b  = wglobal / (NP / 16);
    const int q0 = (wglobal % (NP / 16)) * 16;

    const float* p1b = pos1 + (size_t)b * 3 * NP;
    const float* p2b = pos2 + (size_t)b * 3 * NP;

    // Query tile: A-matrix 16x4 f32 layout — lanes 0-15: K=0,1 ; lanes 16-31: K=2,3(=0)
    const float qx = p1b[q0 + l];
    const float qy = p1b[NP + q0 + l];
    const float qz = p1b[2 * NP + q0 + l];
    const float qq = qx * qx + qy * qy + qz * qz;
    v2f a;
    a.x = half ? qz : qx;
    a.y = half ? 0.f : qy;

    for (int pass = 0; pass < 2; ++pass) {
        const float* rp = (pass == 0) ? p2b : p1b;

        float bd[KS]; int bi[KS];
        #pragma unroll
        for (int s = 0; s < KS; ++s) { bd[s] = 3.0e38f; bi[s] = 0; }

        for (int j0 = 0; j0 < NP; j0 += 32) {
            // own ref point j0+t : norm
            const float ox = rp[j0 + t];
            const float oy = rp[NP + j0 + t];
            const float oz = rp[2 * NP + j0 + t];
            s_rn[wv][t] = ox * ox + oy * oy + oz * oz;

            // B matrices (4x16): same half/K striping as A
            v2f b0, b1;
            if (half == 0) {
                b0.x = ox; b0.y = oy;                 // own point is j0+l
                b1.x = rp[j0 + 16 + l];
                b1.y = rp[NP + j0 + 16 + l];
            } else {
                b0.x = rp[2 * NP + j0 + l]; b0.y = 0.f;
                b1.x = oz;                  b1.y = 0.f;  // own point is j0+16+l
            }

            v8f c0 = {}; v8f c1 = {};
            c0 = __builtin_amdgcn_wmma_f32_16x16x4_f32(false, a, false, b0,
                                                       (short)0, c0, false, false);
            c1 = __builtin_amdgcn_wmma_f32_16x16x4_f32(false, a, false, b1,
                                                       (short)0, c1, false, false);

            // stage accumulators -> LDS so lane r owns row r
            #pragma unroll
            for (int v = 0; v < 8; ++v) {
                s_tile[wv][v + 8 * half][l]      = c0[v];
                s_tile[wv][v + 8 * half][16 + l] = c1[v];
            }
            asm volatile("s_wait_dscnt 0x0" ::: "memory");
            __builtin_amdgcn_wave_barrier();

            // lane owns row=l, columns [half*16, half*16+16)
            #pragma unroll
            for (int n = 0; n < 16; ++n) {
                const int col = half * 16 + n;
                const float d = qq + s_rn[wv][col] - 2.f * s_tile[wv][l][col];
                const int id = j0 + col;
                if (d < bd[KS - 1]) {
                    bd[KS - 1] = d; bi[KS - 1] = id;
                    #pragma unroll
                    for (int s = KS - 1; s > 0; --s) {
                        if (bd[s] < bd[s - 1]) {
                            const float td = bd[s - 1]; bd[s - 1] = bd[s]; bd[s] = td;
                            const int   ti = bi[s - 1]; bi[s - 1] = bi[s]; bi[s] = ti;
                        }
                    }
                }
            }
            asm volatile("s_wait_dscnt 0x0" ::: "memory");
            __builtin_amdgcn_wave_barrier();
        }

        // merge the two sorted half-lists of each row (lane r and lane r+16)
        #pragma unroll
        for (int s = 0; s < KS; ++s) { s_md[wv][t][s] = bd[s]; s_mi[wv][t][s] = bi[s]; }
        asm volatile("s_wait_dscnt 0x0" ::: "memory");
        __builtin_amdgcn_wave_barrier();

        if (t < 16) {
            int ia = 0, ib = 0;
            int* dst = (pass == 0) ? &s_idxI[wv][t][0] : &s_idxS[wv][t][0];
            for (int k = 0; k < KS; ++k) {
                const float da = s_md[wv][t][ia];
                const float db = s_md[wv][t + 16][ib];
                if (da <= db) { dst[k] = s_mi[wv][t][ia]; ++ia; }
                else          { dst[k] = s_mi[wv][t + 16][ib]; ++ib; }
            }
        }
        asm volatile("s_wait_dscnt 0x0" ::: "memory");
        __builtin_amdgcn_wave_barrier();
    }

    // ---- fused grouping / concat ----
    const int fac = *factor_p;                 // runtime scalar input
    const int mk  = l;                         // k slot handled by this lane
    const bool self = (mk < fac);
    const int kk = self ? mk : (mk - fac);
    const float* fsel = (self ? feats1 : feats2) + (size_t)b * CHN * NP;
    const float* psel = self ? p1b : p2b;

    int cols[8];
    #pragma unroll
    for (int i = 0; i < 8; ++i) {
        const int n = 2 * i + half;
        cols[i] = self ? s_idxS[wv][n][kk] : s_idxI[wv][n][kk];
    }

    // new_pos1 [B,3,N,16]
    #pragma unroll
    for (int c3 = 0; c3 < 3; ++c3) {
        #pragma unroll
        for (int i = 0; i < 8; ++i) {
            out_pos[(((size_t)b * 3 + c3) * NP + (q0 + 2 * i + half)) * KS + mk] =
                psel[c3 * NP + cols[i]];
        }
    }
    // new_feats1 [B,C,N,16] — neighbor column fixed across channels
    for (int c = 0; c < CHN; ++c) {
        #pragma unroll
        for (int i = 0; i < 8; ++i) {
            out_feats[(((size_t)b * CHN + c) * NP + (q0 + 2 * i + half)) * KS + mk] =
                fsel[(size_t)c * NP + cols[i]];
        }
    }
}

extern "C" void kernel_launch(void* const* d_in, const int* in_sizes, int n_in,
                              void* d_out, int out_size, void* d_ws, size_t ws_size,
                              hipStream_t stream) {
    const float* pos1   = (const float*)d_in[0];
    const float* pos2   = (const float*)d_in[1];
    const float* feats1 = (const float*)d_in[2];
    const float* feats2 = (const float*)d_in[3];
    const int*   factor = (const int*)d_in[4];

    float* out_pos   = (float*)d_out;
    float* out_feats = out_pos + (size_t)BN * 3 * NP * KS;

    const int totalWaves = BN * (NP / 16);       // 1024 query tiles
    const int blocks = totalWaves / WAVES;       // 256 blocks x 128 threads
    hipLaunchKernelGGL(knn_mix_kernel, dim3(blocks), dim3(WAVES * 32), 0, stream,
                       pos1, pos2, feats1, feats2, factor, out_pos, out_feats);
}